// FeatureFBPConnection_5085241279048
// MI455X (gfx1250) — compile-verified
//
#include <hip/hip_runtime.h>
#include <math.h>

#ifndef M_PI
#define M_PI 3.14159265358979323846
#endif

typedef float v2f __attribute__((ext_vector_type(2)));
typedef float v8f __attribute__((ext_vector_type(8)));

#define Wd   128   // detector columns / filter length
#define NA   128   // angles
#define NV   8     // views
#define NSL  2048  // B*C*H slices
#define SPB  4     // slices per block

// ---------------------------------------------------------------------------
// Build the transposed circulant Ram-Lak filter matrix:
//   HmT[n*128 + k] = h[(n-k) mod 128],  h = real(ifft(|fftfreq(128)|))
// so that filt[r][n] = sum_k raw[r][k] * HmT[n*128+k].
// ---------------------------------------------------------------------------
__global__ void __launch_bounds__(128) build_filter(float* __restrict__ HmT) {
  __shared__ float h[Wd];
  const int n = threadIdx.x;
  double acc = 0.0;
  for (int k = 0; k < Wd; ++k) {
    int f = (k <= Wd / 2) ? k : k - Wd;              // fftfreq * W
    double ramp = (double)(f < 0 ? -f : f) / (double)Wd;
    acc += ramp * cos(2.0 * M_PI * (double)k * (double)n / (double)Wd);
  }
  h[n] = (float)(acc / (double)Wd);                  // ifft normalization
  __syncthreads();
  for (int k = 0; k < Wd; ++k)
    HmT[n * Wd + k] = h[(n - k) & (Wd - 1)];
}

// ---------------------------------------------------------------------------
// Fully fused kernel: one block = 4 slices (b,c,h).
//  phase 1: load 8 raw view rows per slice into LDS (32x128) + angle tables
//  phase 2: WMMA f32 16x16x4 GEMM: filt[32][128] = raw[32][128] x Hm[128][128]
//           (each B tile is reused by two A tiles -> two v_wmma per load)
//  phase 3: back-projection with fused view-axis interpolation; the per-angle
//           geometry (t, taps, weights) is shared across the 4 slices.
// ---------------------------------------------------------------------------
__global__ void __launch_bounds__(256) fbp_kernel(
    const float* __restrict__ f0, const float* __restrict__ f1,
    const float* __restrict__ f2, const float* __restrict__ f3,
    const float* __restrict__ f4, const float* __restrict__ f5,
    const float* __restrict__ f6, const float* __restrict__ f7,
    const float* __restrict__ HmT, float* __restrict__ out) {
  __shared__ float raw [SPB * 8 * Wd];   // rows: slice*8 + view   (16 KB)
  __shared__ float filt[SPB * 8 * Wd];   //                        (16 KB)
  __shared__ float cosT[NA], sinT[NA], wvT[NA];
  __shared__ int   v0T[NA], v1T[NA];

  const int tid    = threadIdx.x;
  const int lane   = tid & 31;
  const int wave   = tid >> 5;          // 8 waves
  const int slice0 = blockIdx.x * SPB;

  // ---- phase 1: raw views -> LDS, and angle tables -------------------------
  {
    const int w   = tid & 127;
    const int ls0 = tid >> 7;           // 0 or 1
#pragma unroll
    for (int dl = 0; dl < SPB / 2; ++dl) {
      const int ls = ls0 + dl * 2;      // local slice 0..3
      const int g  = (slice0 + ls) * Wd + w;
      const int lb = (ls << 3) * Wd + w;
      raw[lb + 0 * Wd] = f0[g];
      raw[lb + 1 * Wd] = f1[g];
      raw[lb + 2 * Wd] = f2[g];
      raw[lb + 3 * Wd] = f3[g];
      raw[lb + 4 * Wd] = f4[g];
      raw[lb + 5 * Wd] = f5[g];
      raw[lb + 6 * Wd] = f6[g];
      raw[lb + 7 * Wd] = f7[g];
    }
  }
  if (tid < NA) {
    const float theta = (float)tid * (float)(M_PI / (double)NA);
    cosT[tid] = cosf(theta);
    sinT[tid] = sinf(theta);
    // half-pixel linear resize V=8 -> A=128 (edge-clamped)
    const float va = ((float)tid + 0.5f) * ((float)NV / (float)NA) - 0.5f;
    const float vf = floorf(va);
    wvT[tid] = va - vf;
    int v0 = (int)vf, v1 = (int)vf + 1;
    v0T[tid] = v0 < 0 ? 0 : (v0 > 7 ? 7 : v0);
    v1T[tid] = v1 < 0 ? 0 : (v1 > 7 ? 7 : v1);
  }
  __syncthreads();

  // ---- phase 2: ramp filter as WMMA f32 16x16x4 GEMM (M=32,N=128,K=128) ----
  {
    v8f accA = {};                      // rows 0..15  (slices 0,1)
    v8f accB = {};                      // rows 16..31 (slices 2,3)
    const int m    = lane & 15;         // A: row M, B: col N (within tile)
    const int koff = (lane >> 4) << 1;  // 0 (lanes 0-15) or 2 (lanes 16-31)
    const float* a0 = raw + m * Wd + koff;                  // A rows 0..15
    const float* a1 = raw + (16 + m) * Wd + koff;           // A rows 16..31
    const float* bp = HmT + (wave * 16 + m) * Wd + koff;    // B[k][col] = HmT[col][k]
#pragma unroll 4
    for (int k0 = 0; k0 < Wd; k0 += 4) {
      v2f Bv  = *(const v2f*)(bp + k0);
      v2f Av0 = *(const v2f*)(a0 + k0);
      v2f Av1 = *(const v2f*)(a1 + k0);
      accA = __builtin_amdgcn_wmma_f32_16x16x4_f32(
          false, Av0, false, Bv, (short)0, accA, false, false);
      accB = __builtin_amdgcn_wmma_f32_16x16x4_f32(
          false, Av1, false, Bv, (short)0, accB, false, false);
    }
    const int rbase = (lane >> 4) << 3; // C/D: row = v + (lane<16 ? 0 : 8)
    const int col   = wave * 16 + m;
#pragma unroll
    for (int v = 0; v < 8; ++v) {
      filt[(rbase + v) * Wd + col]        = accA[v];
      filt[(16 + rbase + v) * Wd + col]   = accB[v];
    }
  }
  __syncthreads();

  // ---- phase 3: back-projection with fused view interpolation --------------
  const float scale = (float)(M_PI / (2.0 * (double)NA));
  for (int i = 0; i < 64; ++i) {
    const int   p = tid + (i << 8);     // pixel index in 128x128 image
    const float X = (float)(p & 127) - 63.5f;
    const float Y = (float)(p >> 7) - 63.5f;
    float acc[SPB] = {0.0f, 0.0f, 0.0f, 0.0f};
    for (int a = 0; a < NA; ++a) {
      float t = fmaf(X, cosT[a], fmaf(Y, sinT[a], 63.5f));
      t = fminf(fmaxf(t, 0.0f), 127.0f);
      const int   t0 = (int)t;          // t >= 0, trunc == floor
      int         t1 = t0 + 1; if (t1 > 127) t1 = 127;
      const float w1 = t - (float)t0;
      const float wv = wvT[a];
      const float* F0 = filt + v0T[a] * Wd;   // view v0 row, slice 0
      const float* F1 = filt + v1T[a] * Wd;   // view v1 row, slice 0
#pragma unroll
      for (int s = 0; s < SPB; ++s) {
        const int off = s << 10;              // +1024 floats per slice
        float a00 = F0[off + t0], a01 = F0[off + t1];
        float a10 = F1[off + t0], a11 = F1[off + t1];
        float c0 = fmaf(wv, a10 - a00, a00);
        float c1 = fmaf(wv, a11 - a01, a01);
        acc[s] += fmaf(w1, c1 - c0, c0);
      }
    }
#pragma unroll
    for (int s = 0; s < SPB; ++s)
      out[(size_t)(slice0 + s) * 16384 + p] = acc[s] * scale;
  }
}

extern "C" void kernel_launch(void* const* d_in, const int* in_sizes, int n_in,
                              void* d_out, int out_size, void* d_ws, size_t ws_size,
                              hipStream_t stream) {
  (void)in_sizes; (void)n_in; (void)out_size; (void)ws_size;
  const float* f0 = (const float*)d_in[0];
  const float* f1 = (const float*)d_in[1];
  const float* f2 = (const float*)d_in[2];
  const float* f3 = (const float*)d_in[3];
  const float* f4 = (const float*)d_in[4];
  const float* f5 = (const float*)d_in[5];
  const float* f6 = (const float*)d_in[6];
  const float* f7 = (const float*)d_in[7];
  float* HmT = (float*)d_ws;              // 128*128 floats = 64 KB
  float* out = (float*)d_out;

  build_filter<<<1, 128, 0, stream>>>(HmT);
  fbp_kernel<<<NSL / SPB, 256, 0, stream>>>(f0, f1, f2, f3, f4, f5, f6, f7, HmT, out);
}